// SparseRandomSampling_4483945857083
// MI455X (gfx1250) — compile-verified
//
#include <hip/hip_runtime.h>
#include <stdint.h>

#define C_CH   128
#define H_IMG  1024
#define W_IMG  1024
#define FH_N   256
#define FW_N   256
#define L_POS  (FH_N * FW_N)      // 65536 positions
#define KSZ    16                 // 4x4 patch
#define HALF_N (L_POS * KSZ / 2)  // 524288: threefry counter pairing distance
#define TILE_FLOATS (16 * W_IMG)  // 16 contiguous image rows = 64KB

typedef unsigned int u32x4 __attribute__((ext_vector_type(4)));
typedef int          i32x4 __attribute__((ext_vector_type(4)));
typedef int          i32x8 __attribute__((ext_vector_type(8)));

#if defined(__has_builtin) && __has_builtin(__builtin_amdgcn_tensor_load_to_lds)
#define HAVE_TDM 1
#else
#define HAVE_TDM 0
#endif

#if HAVE_TDM
// One 64KB contiguous line: tensor_dim0 = tile_dim0 = 16384 f32, 1 row.
__device__ __forceinline__ void tdm_load_tile(const float* src, uint32_t lds_byte) {
  const uint64_t ga = (uint64_t)(uintptr_t)src;

  // D# group 0: count=1(valid), lds_addr, global_addr[56:0], type=2("image")
  u32x4 g0;
  g0[0] = 1u;
  g0[1] = lds_byte;
  g0[2] = (uint32_t)ga;
  g0[3] = ((uint32_t)(ga >> 32) & 0x01FFFFFFu) | (2u << 30);

  // D# group 1: wg_mask=0, data_size=2(4B), tensor_dim0=tile_dim0=16384,
  // tensor_dim1=tile_dim1=1, tensor_dim0_stride=16384
  i32x8 g1;
  g1[0] = (int)(2u << 16);
  g1[1] = (int)((uint32_t)TILE_FLOATS << 16);   // tensor_dim0[15:0] @ bits 63:48
  g1[2] = (int)(1u << 16);                      // tensor_dim1[15:0] @ bits 95:80
  g1[3] = (int)((uint32_t)TILE_FLOATS << 16);   // tile_dim0 @ bits 127:112
  g1[4] = 1;                                    // tile_dim1 = 1
  g1[5] = (int)TILE_FLOATS;                     // tensor_dim0_stride[31:0]
  g1[6] = 0;
  g1[7] = 0;

  i32x4 gz = {0, 0, 0, 0};
#if __clang_major__ >= 23
  i32x8 gz8 = {0, 0, 0, 0, 0, 0, 0, 0};
  __builtin_amdgcn_tensor_load_to_lds(g0, g1, gz, gz, gz8, 0);
#else
  __builtin_amdgcn_tensor_load_to_lds(g0, g1, gz, gz, 0);
#endif
}

template <int N>
__device__ __forceinline__ void tdm_wait() {
#if __has_builtin(__builtin_amdgcn_s_wait_tensorcnt)
  __builtin_amdgcn_s_wait_tensorcnt(N);          // N is a constant expression here
#else
  asm volatile("s_wait_tensorcnt %0" :: "i"(N) : "memory");
#endif
}
#endif  // HAVE_TDM

// ---------------- gather via double-buffered TDM-staged LDS tiles -------------------------
// Block: (fhb, cp) -> image rows [16*fhb, 16*fhb+16) for channels c0=2*cp, c1=2*cp+1.
// Two 64KB tiles ping-pong in LDS; tile B's DMA overlaps tile A's gather.
__global__ void __launch_bounds__(256) srs_gather(const float* __restrict__ x,
                                                  const uint32_t* __restrict__ idx,
                                                  float* __restrict__ out) {
  __shared__ float tile[2][TILE_FLOATS];           // 128 KB of the 320 KB WGP LDS
  const uint32_t fhb = blockIdx.x;                 // 0..63  (block of 4 fh rows)
  const uint32_t c0  = blockIdx.y * 2u;            // 0,2,..,126
  const uint32_t tid = threadIdx.x;

  const float* src0 = x + (size_t)c0 * (H_IMG * W_IMG) + (size_t)fhb * TILE_FLOATS;
  const float* src1 = src0 + (size_t)(H_IMG * W_IMG);

#if HAVE_TDM
  if (tid < 32) {                                  // one wave issues both DMAs
    tdm_load_tile(src0, (uint32_t)(uintptr_t)&tile[0][0]);
    tdm_load_tile(src1, (uint32_t)(uintptr_t)&tile[1][0]);
    tdm_wait<1>();                                 // tile 0 landed; tile 1 in flight
  }
#else
  {
    const float4* s4 = (const float4*)src0;
    float4* d4 = (float4*)&tile[0][0];
    for (uint32_t i = tid; i < TILE_FLOATS / 4; i += 256u) d4[i] = s4[i];
  }
#endif
  __syncthreads();

  // 1024 positions per block: L = fhb*1024 + p, p = fh_local*256 + fw
  const uint32_t Lbase = fhb * 1024u;
  uint32_t kk[4], off[4];
#pragma unroll
  for (uint32_t i = 0; i < 4; ++i) {
    const uint32_t p  = tid + i * 256u;            // 0..1023
    const uint32_t fl = p >> 8;                    // fh within block: 0..3
    const uint32_t fw = p & 255u;
    kk[i] = idx[Lbase + p];                        // 0..15, channel-independent
    const uint32_t r = kk[i] >> 2, cc = kk[i] & 3u;
    off[i] = (fl * 4u + r) * W_IMG + fw * 4u + cc; // LDS gather offset (floats)
  }

#pragma unroll
  for (uint32_t i = 0; i < 4; ++i)                 // channel c0 from tile 0
    out[(size_t)c0 * L_POS + Lbase + tid + i * 256u] = tile[0][off[i]];

#if HAVE_TDM
  if (tid < 32) tdm_wait<0>();                     // tile 1 landed
#else
  __syncthreads();
  {
    const float4* s4 = (const float4*)src1;
    float4* d4 = (float4*)&tile[1][0];
    for (uint32_t i = tid; i < TILE_FLOATS / 4; i += 256u) d4[i] = s4[i];
  }
#endif
  __syncthreads();

#pragma unroll
  for (uint32_t i = 0; i < 4; ++i)                 // channel c0+1 from tile 1
    out[(size_t)(c0 + 1u) * L_POS + Lbase + tid + i * 256u] = tile[1][off[i]];
}

// ---------------- threefry2x32 (JAX default PRNG, key = PRNGKey(42) = {0,42}) -------------
__device__ __forceinline__ uint32_t rotl32(uint32_t x, int r) {
  return (x << r) | (x >> (32 - r));
}

__device__ __forceinline__ void threefry2x32(uint32_t x0, uint32_t x1,
                                             uint32_t& o0, uint32_t& o1) {
  const uint32_t k0 = 0u, k1 = 42u;
  const uint32_t k2 = k0 ^ k1 ^ 0x1BD11BDAu;
  uint32_t X0 = x0 + k0, X1 = x1 + k1;
#define TF_R(r) { X0 += X1; X1 = rotl32(X1, (r)); X1 ^= X0; }
  TF_R(13) TF_R(15) TF_R(26) TF_R(6)
  X0 += k1; X1 += k2 + 1u;
  TF_R(17) TF_R(29) TF_R(16) TF_R(24)
  X0 += k2; X1 += k0 + 2u;
  TF_R(13) TF_R(15) TF_R(26) TF_R(6)
  X0 += k0; X1 += k1 + 3u;
  TF_R(17) TF_R(29) TF_R(16) TF_R(24)
  X0 += k1; X1 += k2 + 4u;
  TF_R(13) TF_R(15) TF_R(26) TF_R(6)
  X0 += k2; X1 += k0 + 5u;
#undef TF_R
  o0 = X0; o1 = X1;
}

// Thread t < 32768 serves positions L=t and L=t+32768: counters t*16+j pair with
// t*16+j+524288 (JAX splits the flat iota in halves), so (o0 -> L=t, o1 -> L=t+32768).
// Stable argmin of (bits>>9) == stable argmin of jax.random.uniform + argsort[...,:1].
__global__ void __launch_bounds__(256) srs_indices(uint32_t* __restrict__ idx) {
  const uint32_t t = blockIdx.x * blockDim.x + threadIdx.x;   // 0..32767
  const uint32_t base = t * KSZ;
  uint32_t best0 = 0xFFFFFFFFu, best1 = 0xFFFFFFFFu;
  uint32_t j0 = 0, j1 = 0;
#pragma unroll
  for (uint32_t j = 0; j < KSZ; ++j) {
    uint32_t o0, o1;
    threefry2x32(base + j, base + j + HALF_N, o0, o1);
    const uint32_t m0 = o0 >> 9, m1 = o1 >> 9;   // 23-bit mantissa keys
    if (m0 < best0) { best0 = m0; j0 = j; }
    if (m1 < best1) { best1 = m1; j1 = j; }
  }
  idx[t] = j0;
  idx[t + L_POS / 2] = j1;
}

extern "C" void kernel_launch(void* const* d_in, const int* in_sizes, int n_in,
                              void* d_out, int out_size, void* d_ws, size_t ws_size,
                              hipStream_t stream) {
  (void)in_sizes; (void)n_in; (void)out_size; (void)ws_size;
  const float* x  = (const float*)d_in[0];
  float* out      = (float*)d_out;
  uint32_t* idx   = (uint32_t*)d_ws;               // 65536 * 4B = 256 KB scratch

  srs_indices<<<dim3(L_POS / 2 / 256), dim3(256), 0, stream>>>(idx);
  srs_gather<<<dim3(64, C_CH / 2), dim3(256), 0, stream>>>(x, idx, out);
}